// GraphPerformanceTrace_54107997995650
// MI455X (gfx1250) — compile-verified
//
#include <hip/hip_runtime.h>
#include <hip/hip_bf16.h>

typedef __attribute__((ext_vector_type(2))) float v2f;
typedef __attribute__((ext_vector_type(8))) float v8f;

#define NN 1024
#define BS 32
#define LL 32
#define TT 64
#define STEPS 63
#define JSPLIT 8
#define JCHUNK (NN / JSPLIT)   // 128 rows per block

// ---------------------------------------------------------------------------
// Kernel P: base = user_b @ node_rec.T ; nlf_new = user_f @ node_send.T
// GEMM M=32 (bs), N=1024, K=32 via V_WMMA_F32_16X16X4_F32 (K chunked by 4).
// One wave = one 16x16 output tile. grid: 32 blocks x 8 waves = 256 jobs.
// (verified: emits 8 chained v_wmma_f32_16x16x4_f32 per wave)
// ---------------------------------------------------------------------------
__global__ __launch_bounds__(256) void proj_wmma(
    const float* __restrict__ ub, const float* __restrict__ uf,
    const float* __restrict__ node_emb,   // (N, 2L): send=[:, :L], rec=[:, L:]
    float* __restrict__ base, float* __restrict__ nlf_new)
{
  int wave = threadIdx.x >> 5;
  int lane = threadIdx.x & 31;
  int job  = blockIdx.x * 8 + wave;      // 0..255
  int mat  = job >> 7;                   // 0: base(ub,rec)  1: nlf(uf,send)
  int tile = job & 127;
  int mt   = tile >> 6;                  // 0..1   (bs tiles)
  int nt   = tile & 63;                  // 0..63  (N tiles)

  const float* A = mat ? uf : ub;        // (32, 32) row-major
  const int embOff = mat ? 0 : LL;       // send vs rec columns
  float* D = mat ? nlf_new : base;       // (32, N) row-major

  int half = lane >> 4;                  // 0 | 1
  int l16  = lane & 15;

  v8f c = {0.f,0.f,0.f,0.f,0.f,0.f,0.f,0.f};
  int m = mt * 16 + l16;
  int n = nt * 16 + l16;
  #pragma unroll
  for (int kc = 0; kc < 8; ++kc) {
    int ka = kc * 4 + half * 2;          // K element pair per half-wave
    v2f a, b;
    a.x = A[m * LL + ka];
    a.y = A[m * LL + ka + 1];
    b.x = node_emb[n * (2 * LL) + embOff + ka];
    b.y = node_emb[n * (2 * LL) + embOff + ka + 1];
    c = __builtin_amdgcn_wmma_f32_16x16x4_f32(
        false, a, false, b, (short)0, c, false, false);
  }
  int ncol = nt * 16 + l16;
  #pragma unroll
  for (int i = 0; i < 8; ++i) {
    int mrow = mt * 16 + half * 8 + i;   // C/D layout: VGPR i -> M=i / M=i+8
    D[mrow * NN + ncol] = c[i];
  }
}

// ---------------------------------------------------------------------------
// Kernel S1: partial GEMV  partial[b][js][k] = sum_{j in slice} lp[j]*adj[b,j,k]
// grid (JSPLIT=8, bs=32) = 256 blocks x 256 threads.
// Each thread owns 4 consecutive columns -> global_load_b128 inner loop.
// lp row-slice broadcast via LDS. adj resident in 192MB L2 after step 0.
// ---------------------------------------------------------------------------
__global__ __launch_bounds__(256) void spatial_partial(
    const float* __restrict__ adj,       // (bs, N, N)
    const float* __restrict__ lp,        // (bs, N)
    float* __restrict__ partial)         // (bs, JSPLIT, N)
{
  __shared__ float slp[JCHUNK];
  const int b  = blockIdx.y;
  const int js = blockIdx.x;
  const int j0 = js * JCHUNK;

  if (threadIdx.x < JCHUNK)
    slp[threadIdx.x] = lp[b * NN + j0 + threadIdx.x];
  __syncthreads();

  const int k4 = threadIdx.x * 4;       // 4 consecutive columns per thread
  const float* __restrict__ abase = adj + (size_t)b * NN * NN + (size_t)j0 * NN + k4;

  float ax = 0.f, ay = 0.f, az = 0.f, aw = 0.f;
  #pragma unroll 4
  for (int j = 0; j < JCHUNK; ++j) {
    const float4 v = *reinterpret_cast<const float4*>(abase + (size_t)j * NN);
    const float s = slp[j];
    ax = fmaf(s, v.x, ax);
    ay = fmaf(s, v.y, ay);
    az = fmaf(s, v.z, az);
    aw = fmaf(s, v.w, aw);
  }
  float4 out; out.x = ax; out.y = ay; out.z = az; out.w = aw;
  *reinterpret_cast<float4*>(partial + (size_t)b * JSPLIT * NN + (size_t)js * NN + k4) = out;
}

// ---------------------------------------------------------------------------
// Kernel S2: reduce partials + temporal + sigmoid + pred gather.
// grid 128 blocks x 256 threads; one (b,k) per thread.
// ---------------------------------------------------------------------------
__global__ __launch_bounds__(256) void step_combine(
    const float* __restrict__ partial,   // (bs, JSPLIT, N)
    const float* __restrict__ lp,        // (bs, N)
    const float* __restrict__ nlf,       // (bs, N)
    const float* __restrict__ base,      // (bs, N)
    const int* __restrict__ time_seq,    // (bs, T)
    const int* __restrict__ skill_seq,   // (bs, T)
    float* __restrict__ cur,             // (bs, N) out
    float* __restrict__ preds, int t)
{
  const int idx = blockIdx.x * blockDim.x + threadIdx.x;  // 0..bs*N-1
  const int b = idx >> 10;
  const int k = idx & (NN - 1);

  float acc = 0.f;
  #pragma unroll
  for (int s = 0; s < JSPLIT; ++s)
    acc += partial[(size_t)b * JSPLIT * NN + (size_t)s * NN + k];

  const float lpk = lp[b * NN + k];
  float dt = fabsf((float)(time_seq[b * TT + t + 1] - time_seq[b * TT + t])) + 1e-10f;
  float temporal = __expf(-dt * 0.1f * nlf[b * NN + k]) * lpk;
  float spatial  = lpk * (1.0f / (2.0f * NN)) + acc * (1.0f / (2.0f * NN * NN));
  float x = base[b * NN + k] + temporal + spatial;
  float s = 1.0f / (1.0f + __expf(-x));
  cur[b * NN + k] = s;
  if (k == skill_seq[b * TT + t + 1])
    preds[t * BS + b] = s;
}

// ---------------------------------------------------------------------------
// Kernel U: conv1(5-tap) -> relu -> conv2 -> relu -> tanh heads.
// 1 block x 1024 threads: thread (b,o).
// ---------------------------------------------------------------------------
__global__ __launch_bounds__(1024) void update_user(
    const float* __restrict__ ub, const float* __restrict__ uf,
    const float* __restrict__ node_emb,
    const int* __restrict__ skill_seq, const int* __restrict__ label_seq,
    const float* __restrict__ c1w, const float* __restrict__ c1b,
    const float* __restrict__ c2w, const float* __restrict__ c2b,
    const float* __restrict__ fw,  const float* __restrict__ fb,
    const float* __restrict__ bw,  const float* __restrict__ bb,
    float* __restrict__ ub_new, float* __restrict__ uf_new, int t)
{
  __shared__ float h1[BS][LL];
  __shared__ float h2[BS][LL];
  const int tid = threadIdx.x;
  const int b = tid >> 5;
  const int o = tid & 31;
  const int sk = skill_seq[b * TT + t + 1];
  const float lab = (float)label_seq[b * TT + t + 1];

  float acc = c1b[o];
  #pragma unroll 4
  for (int i = 0; i < LL; ++i) {
    const float* w = c1w + o * LL * 5 + i * 5;
    float send = node_emb[sk * (2 * LL) + i];
    float rec  = node_emb[sk * (2 * LL) + LL + i];
    acc += ub[b * LL + i] * w[0] + uf[b * LL + i] * w[1]
         + lab * w[2] + rec * w[3] + send * w[4];
  }
  h1[b][o] = fmaxf(acc, 0.f);
  __syncthreads();

  float acc2 = c2b[o];
  #pragma unroll 4
  for (int i = 0; i < LL; ++i) acc2 += h1[b][i] * c2w[o * LL + i];
  h2[b][o] = fmaxf(acc2, 0.f);
  __syncthreads();

  float af = fb[o], ab2 = bb[o];
  #pragma unroll 4
  for (int i = 0; i < LL; ++i) {
    af  += h2[b][i] * fw[o * LL + i];
    ab2 += h2[b][i] * bw[o * LL + i];
  }
  uf_new[b * LL + o] = tanhf(af);
  ub_new[b * LL + o] = tanhf(ab2);
}

// Zero initial state each call (graph-replay safe).
__global__ __launch_bounds__(256) void init_state(float* __restrict__ ws, int n)
{
  int i = blockIdx.x * blockDim.x + threadIdx.x;
  if (i < n) ws[i] = 0.f;
}

extern "C" void kernel_launch(void* const* d_in, const int* in_sizes, int n_in,
                              void* d_out, int out_size, void* d_ws, size_t ws_size,
                              hipStream_t stream) {
  const float* adj      = (const float*)d_in[0];   // (1, bs, N, N)
  const float* node_emb = (const float*)d_in[1];   // (1, N, 2L)
  const int*   skill    = (const int*)d_in[2];     // (bs, T)
  const int*   times    = (const int*)d_in[3];     // (bs, T)
  const int*   labels   = (const int*)d_in[4];     // (bs, T)
  const float* c1w = (const float*)d_in[5];
  const float* c1b = (const float*)d_in[6];
  const float* c2w = (const float*)d_in[7];
  const float* c2b = (const float*)d_in[8];
  const float* fw  = (const float*)d_in[9];
  const float* fb  = (const float*)d_in[10];
  const float* bw  = (const float*)d_in[11];
  const float* bb  = (const float*)d_in[12];
  float* preds = (float*)d_out;                    // (63, bs, 1)

  float* ws = (float*)d_ws;
  const int PN = BS * NN;                          // 32768
  float* perf[2] = { ws,            ws + PN };
  float* nlf[2]  = { ws + 2 * PN,   ws + 3 * PN };
  float* basebuf = ws + 4 * PN;
  float* partial = ws + 5 * PN;                    // bs * JSPLIT * N = 8*PN floats
  float* ubuf    = ws + 13 * PN;                   // ub0, ub1, uf0, uf1 (1024 each)
  float* ub[2]   = { ubuf,          ubuf + 1024 };
  float* uf[2]   = { ubuf + 2048,   ubuf + 3072 };

  // zero perf/nlf double buffers and user-state buffers
  init_state<<<(4 * PN + 255) / 256, 256, 0, stream>>>(ws, 4 * PN);
  init_state<<<(4096 + 255) / 256, 256, 0, stream>>>(ubuf, 4096);

  for (int t = 0; t < STEPS; ++t) {
    const int pi = t & 1, po = pi ^ 1;
    // projections with previous user state -> base, nlf[po]
    proj_wmma<<<32, 256, 0, stream>>>(ub[pi], uf[pi], node_emb, basebuf, nlf[po]);
    // phase 1: partial GEMV over adj (256 streaming blocks, b128 loads)
    dim3 gp(JSPLIT, BS);
    spatial_partial<<<gp, 256, 0, stream>>>(adj, perf[pi], partial);
    // phase 2: reduce + temporal + sigmoid + pred
    step_combine<<<(BS * NN) / 256, 256, 0, stream>>>(partial, perf[pi], nlf[pi],
                                                      basebuf, times, skill,
                                                      perf[po], preds, t);
    // user state MLP update
    update_user<<<1, 1024, 0, stream>>>(ub[pi], uf[pi], node_emb, skill, labels,
                                        c1w, c1b, c2w, c2b, fw, fb, bw, bb,
                                        ub[po], uf[po], t);
  }
}